// NeighbourCovariance_38285338476773
// MI455X (gfx1250) — compile-verified
//
#include <hip/hip_runtime.h>

// CDNA5 / gfx1250: wave32, WMMA f32 16x16x4 (f32 A/B -> matches fp32 reference precision)
typedef float v2f __attribute__((ext_vector_type(2)));
typedef float v8f __attribute__((ext_vector_type(8)));

#define NC_EPS 1e-6f
#define K_NB   32
#define F_DIM  16
#define SF     17   // LDS stride for 32x16 staging tiles (conflict-free)
#define SM     20   // LDS stride for 16x16 result tile  (conflict-free)

__global__ __launch_bounds__(32)
void NeighbourCovariance_wmma_kernel(const float* __restrict__ coords,
                                     const float* __restrict__ feats,
                                     const long long* __restrict__ nidx,
                                     float* __restrict__ out,
                                     int V)
{
    __shared__ float ldsF[K_NB * SF];   // feats staged as [k][f]
    __shared__ float ldsG[K_NB * SF];   // moments staged as [k][j]
    __shared__ float ldsM[F_DIM * SM];  // result M[f][j]

    const int v = blockIdx.x;
    if (v >= V) return;                 // block-uniform: EXEC stays all-ones

    const int lane = threadIdx.x;       // 0..31, lane == neighbor slot k
    const int half = lane >> 4;
    const int col  = lane & 15;

    // ---- gather: lane k loads neighbor k's coord (12B) and feature row (64B) ----
    const long long idx = nidx[(size_t)v * K_NB + lane];
    const float* cp = coords + (size_t)idx * 3;
    const float x = cp[0], y = cp[1], z = cp[2];

    const float4* fp = (const float4*)(feats + (size_t)idx * F_DIM);
    const float4 f0 = fp[0], f1 = fp[1], f2 = fp[2], f3 = fp[3];

    // ---- moments G[k][0..15] = [x,y,z, xx,xy,yy, xz,yz,zz, 1, 0...] ----
    float g[16];
    g[0] = x;     g[1] = y;     g[2] = z;
    g[3] = x * x; g[4] = x * y; g[5] = y * y;
    g[6] = x * z; g[7] = y * z; g[8] = z * z;
    g[9] = 1.0f;
    g[10] = 0.0f; g[11] = 0.0f; g[12] = 0.0f; g[13] = 0.0f; g[14] = 0.0f; g[15] = 0.0f;

    float fv[16] = { f0.x, f0.y, f0.z, f0.w,
                     f1.x, f1.y, f1.z, f1.w,
                     f2.x, f2.y, f2.z, f2.w,
                     f3.x, f3.y, f3.z, f3.w };

#pragma unroll
    for (int j = 0; j < 16; ++j) {
        ldsG[lane * SF + j] = g[j];
        ldsF[lane * SF + j] = fv[j];
    }
    __syncthreads();   // single-wave workgroup: barrier -> S_NOP, keeps DS waits

    // ---- M(16x16) = feats^T(16x32) x G(32x16) via 8x v_wmma_f32_16x16x4_f32 ----
    // A lane layout (16x4 f32): lanes 0-15 M=0..15, VGPR0/1 = K=k0,k0+1;
    //                           lanes 16-31          VGPR0/1 = K=k0+2,k0+3.
    // B lane layout (4x16 f32): mirror with N across lanes.
    // A[m][k] = feats[k][m], B[k][n] = G[k][n] -> identical index expression.
    v8f acc = {};
#pragma unroll
    for (int s = 0; s < 8; ++s) {
        const int krow = 4 * s + 2 * half;
        v2f a, b;
        a.x = ldsF[krow * SF + col];
        a.y = ldsF[(krow + 1) * SF + col];
        b.x = ldsG[krow * SF + col];
        b.y = ldsG[(krow + 1) * SF + col];
        acc = __builtin_amdgcn_wmma_f32_16x16x4_f32(
            /*neg_a=*/false, a, /*neg_b=*/false, b,
            /*c_mod=*/(short)0, acc, /*reuse_a=*/false, /*reuse_b=*/false);
    }

    // D layout: lane L holds column n=L&15 for rows M = r + 8*(L>>4), r=0..7
#pragma unroll
    for (int r = 0; r < 8; ++r)
        ldsM[(r + 8 * half) * SM + col] = acc[r];
    __syncthreads();

    // ---- finalize: lane (half,f=col): half0 -> 6 cov entries, half1 -> 3 means ----
    const int f = col;
    float m[10];
#pragma unroll
    for (int j = 0; j < 10; ++j) m[j] = ldsM[f * SM + j];

    const float inv = 1.0f / (m[9] + NC_EPS);
    const float mx = m[0] * inv, my = m[1] * inv, mz = m[2] * inv;

    float* o = out + (size_t)v * 144;
    if (half == 0) {
        // tril order: (0,0) (1,0) (1,1) (2,0) (2,1) (2,2)
        o[f * 6 + 0] = m[3] * inv - mx * mx;
        o[f * 6 + 1] = m[4] * inv - my * mx;
        o[f * 6 + 2] = m[5] * inv - my * my;
        o[f * 6 + 3] = m[6] * inv - mz * mx;
        o[f * 6 + 4] = m[7] * inv - mz * my;
        o[f * 6 + 5] = m[8] * inv - mz * mz;
    } else {
        o[96 + f * 3 + 0] = mx;
        o[96 + f * 3 + 1] = my;
        o[96 + f * 3 + 2] = mz;
    }
}

extern "C" void kernel_launch(void* const* d_in, const int* in_sizes, int n_in,
                              void* d_out, int out_size, void* d_ws, size_t ws_size,
                              hipStream_t stream) {
    (void)n_in; (void)out_size; (void)d_ws; (void)ws_size;
    const float*     coords = (const float*)d_in[0];      // (V,3) f32
    const float*     feats  = (const float*)d_in[1];      // (V,16) f32
    const long long* nidx   = (const long long*)d_in[2];  // (V,32) i64
    float* out = (float*)d_out;                           // (V,144) f32

    const int V = in_sizes[0] / 3;
    dim3 grid((unsigned)V), block(32);
    NeighbourCovariance_wmma_kernel<<<grid, block, 0, stream>>>(coords, feats, nidx, out, V);
}